// GtNet_84267258347956
// MI455X (gfx1250) — compile-verified
//
#include <hip/hip_runtime.h>
#include <stdint.h>

namespace {
constexpr int kB = 16, kH = 256, kW = 256, kHW = kH * kW;
constexpr int kNC = 25;              // 5x5 displacement classes
constexpr int TILE = 32, HALO = 36;  // 32x32 outputs + 2-halo
constexpr int NHALO = HALO * HALO;   // 1296
constexpr int NTHR = 256;            // 8 wave32 waves

// output tuple layout (flat, in return order)
constexpr size_t P_PRED = 0;
constexpr size_t P_MMF  = (size_t)3 * kB * kHW;              //  3,145,728
constexpr size_t P_ATF  = P_MMF + (size_t)kNC * kB * kHW;    // 29,360,128
constexpr size_t P_MMB  = P_ATF + (size_t)kB * kHW;          // 30,408,704
constexpr size_t P_ATB  = P_MMB + (size_t)kNC * kB * kHW;    // 56,623,104
}

typedef unsigned int v4u_t __attribute__((ext_vector_type(4)));
typedef int          v8i_t __attribute__((ext_vector_type(8)));
typedef int          v4i_t __attribute__((ext_vector_type(4)));

// TDM 2D tile load: 36x36 x 4B elements, global row stride 256 elems -> LDS.
// D# layout per CDNA5 ISA.
__device__ __forceinline__ void tdm_load_tile36(unsigned lds_off, const int* gsrc) {
  const unsigned long long ga = (unsigned long long)(uintptr_t)gsrc;
  v4u_t g0;
  g0[0] = 1u;                                   // count=1, user descriptor
  g0[1] = lds_off;                              // lds_addr (bytes)
  g0[2] = (unsigned)ga;                         // global_addr[31:0]
  g0[3] = (unsigned)((ga >> 32) & 0x01ffffffu)  // global_addr[56:32]
        | (2u << 30);                           // type = 2 ("image")
  v8i_t g1;
  g1[0] = (int)(2u << 16);    // data_size=4B; mask/pad/iterate = 0
  g1[1] = (int)(256u << 16);  // [63:48]  tensor_dim0[15:0] = 256
  g1[2] = (int)(256u << 16);  // [95:80]  tensor_dim1[15:0] = 256
  g1[3] = (int)(36u << 16);   // [127:112] tile_dim0 = 36
  g1[4] = 36;                 // [143:128] tile_dim1 = 36; tile_dim2 = 0 (2D)
  g1[5] = 256;                // [191:160] tensor_dim0_stride = 256 elems
  g1[6] = 0;                  // stride0 hi = 0; stride1[15:0] = 0
  g1[7] = 1;                  // [255:224] tensor_dim1_stride[47:16] -> 65536
  v4i_t z4 = {};
#if defined(__clang_major__) && (__clang_major__ >= 23)
  v8i_t z8 = {};
  __builtin_amdgcn_tensor_load_to_lds(g0, g1, z4, z4, z8, 0);
#else
  __builtin_amdgcn_tensor_load_to_lds(g0, g1, z4, z4, 0);
#endif
}

__global__ __launch_bounds__(NTHR)
void gtnet_fused(const float* __restrict__ imF, const float* __restrict__ imB,
                 const int* __restrict__ gmF, const int* __restrict__ gmB,
                 float* __restrict__ out)
{
  __shared__ int s_labF[NHALO];
  __shared__ int s_labB[NHALO];
  __shared__ unsigned s_mF[TILE * TILE];  // per-output 25-bit match bitmaps
  __shared__ unsigned s_mB[TILE * TILE];

  const int t  = threadIdx.x;
  const int x0 = blockIdx.x * TILE;
  const int y0 = blockIdx.y * TILE;
  const int b  = blockIdx.z;

  for (int i = t; i < TILE * TILE; i += NTHR) { s_mF[i] = 0u; s_mB[i] = 0u; }

  const int* gF = gmF + (size_t)b * kHW;
  const int* gB = gmB + (size_t)b * kHW;

  // ---- stage 36x36 label halo tiles into LDS ----
  const bool interior = (x0 >= 2) && (x0 + TILE + 2 <= kW) &&
                        (y0 >= 2) && (y0 + TILE + 2 <= kH);
  if (interior) {
    // Whole halo is in-image: one TDM DMA per plane, issued by wave 0 only
    // (TDM ignores EXEC; per-wave issue would duplicate the DMA 8x).
    if (t < 32) {
      tdm_load_tile36((unsigned)(uintptr_t)&s_labF[0],
                      gF + (y0 - 2) * kW + (x0 - 2));
      tdm_load_tile36((unsigned)(uintptr_t)&s_labB[0],
                      gB + (y0 - 2) * kW + (x0 - 2));
      __builtin_amdgcn_s_wait_tensorcnt(0);
    }
  } else {
    // Boundary: per-lane async copies; OOB slots get -1 sentinel (a padded
    // zero label would alias class 0, so TDM zero-fill is not usable here).
    for (int idx = t; idx < NHALO; idx += NTHR) {
      const int r  = idx / HALO;
      const int c  = idx - r * HALO;
      const int sy = y0 + r - 2;
      const int sx = x0 + c - 2;
      if ((unsigned)sy < (unsigned)kH && (unsigned)sx < (unsigned)kW) {
        const int voff = (sy * kW + sx) * 4;  // byte offset, GVS mode
        const unsigned ldsF = (unsigned)(uintptr_t)&s_labF[idx];
        const unsigned ldsB = (unsigned)(uintptr_t)&s_labB[idx];
        asm volatile("global_load_async_to_lds_b32 %0, %1, %2"
                     :: "v"(ldsF), "v"(voff), "s"(gF) : "memory");
        asm volatile("global_load_async_to_lds_b32 %0, %1, %2"
                     :: "v"(ldsB), "v"(voff), "s"(gB) : "memory");
      } else {
        s_labF[idx] = -1;
        s_labB[idx] = -1;
      }
    }
    asm volatile("s_wait_asynccnt 0x0" ::: "memory");
  }
  __syncthreads();

  // ---- scatter: each in-bounds source sets bit j in its target's bitmap ----
  for (int idx = t; idx < NHALO; idx += NTHR) {
    const int r = idx / HALO;
    const int c = idx - r * HALO;
    const int jf = s_labF[idx];
    if (jf >= 0) {
      const int dy = jf / 5, dx = jf - 5 * dy;
      const int ty = r - dy, tx = c - dx;   // target in tile-interior coords
      if ((unsigned)ty < (unsigned)TILE && (unsigned)tx < (unsigned)TILE)
        atomicOr(&s_mF[ty * TILE + tx], 1u << jf);   // ds_or_b32, order-free
    }
    const int jb = s_labB[idx];
    if (jb >= 0) {
      const int dy = jb / 5, dx = jb - 5 * dy;
      const int ty = r - dy, tx = c - dx;
      if ((unsigned)ty < (unsigned)TILE && (unsigned)tx < (unsigned)TILE)
        atomicOr(&s_mB[ty * TILE + tx], 1u << jb);
    }
  }
  __syncthreads();

  // ---- output: each thread owns a 1x4 x-strip (b128 stores) ----
  const int ty  = t >> 3;
  const int tx4 = (t & 7) << 2;
  const int oy  = y0 + ty;
  const int ox  = x0 + tx4;
  const size_t pix = (size_t)oy * kW + ox;

  const float* iF = imF + ((size_t)b * 6 + 3) * kHW;  // im_input_f[:, -3:]
  const float* iB = imB + ((size_t)b * 6 + 3) * kHW;

  float segf[4], pf[3][4], pb[3][4];
#pragma unroll
  for (int i = 0; i < 4; ++i) {
    unsigned mF = s_mF[ty * TILE + tx4 + i];
    unsigned mB = s_mB[ty * TILE + tx4 + i];
    segf[i] = (float)__popc(mF);
    float a0 = 0.f, a1 = 0.f, a2 = 0.f;
    while (mF) {  // avg ~1 iteration; fixed bit order -> deterministic sums
      const int j = __builtin_ctz(mF); mF &= mF - 1u;
      const int sp = (oy + j / 5 - 2) * kW + (ox + i + (j - 5 * (j / 5)) - 2);
      a0 += iF[sp]; a1 += iF[kHW + sp]; a2 += iF[2 * kHW + sp];
    }
    pf[0][i] = a0; pf[1][i] = a1; pf[2][i] = a2;
    float b0 = 0.f, b1 = 0.f, b2 = 0.f;
    while (mB) {
      const int j = __builtin_ctz(mB); mB &= mB - 1u;
      const int sp = (oy + j / 5 - 2) * kW + (ox + i + (j - 5 * (j / 5)) - 2);
      b0 += iB[sp]; b1 += iB[kHW + sp]; b2 += iB[2 * kHW + sp];
    }
    pb[0][i] = b0; pb[1][i] = b1; pb[2][i] = b2;
  }

  float af[4], ab4[4];
#pragma unroll
  for (int i = 0; i < 4; ++i) {
    const float r1 = 1.0f - segf[i];
    ab4[i] = r1 > 0.0f ? r1 : 0.0f;   // attn_b = relu(1 - seg_f)
    af[i]  = 1.0f - ab4[i];           // attn_f
  }

#pragma unroll
  for (int cch = 0; cch < 3; ++cch) {
    float4 v = make_float4(af[0] * pf[cch][0] + ab4[0] * pb[cch][0],
                           af[1] * pf[cch][1] + ab4[1] * pb[cch][1],
                           af[2] * pf[cch][2] + ab4[2] * pb[cch][2],
                           af[3] * pf[cch][3] + ab4[3] * pb[cch][3]);
    *reinterpret_cast<float4*>(out + P_PRED + (size_t)(b * 3 + cch) * kHW + pix) = v;
  }

  *reinterpret_cast<float4*>(out + P_ATF + (size_t)b * kHW + pix) =
      make_float4(af[0], af[1], af[2], af[3]);
  *reinterpret_cast<float4*>(out + P_ATB + (size_t)b * kHW + pix) =
      make_float4(ab4[0], ab4[1], ab4[2], ab4[3]);

  // ---- one-hot masks: zero-fill quads (no compares), then poke the single
  //      hot element per pixel. Same-wave stores complete in order. ----
  int lf[4], lb[4];
#pragma unroll
  for (int i = 0; i < 4; ++i) {
    lf[i] = s_labF[(ty + 2) * HALO + (tx4 + 2 + i)];
    lb[i] = s_labB[(ty + 2) * HALO + (tx4 + 2 + i)];
  }
  float* mfp = out + P_MMF + (size_t)b * kNC * kHW + pix;
  float* mbp = out + P_MMB + (size_t)b * kNC * kHW + pix;
  const float4 zero4 = make_float4(0.f, 0.f, 0.f, 0.f);
#pragma unroll
  for (int j = 0; j < kNC; ++j) {
    *reinterpret_cast<float4*>(mfp + (size_t)j * kHW) = zero4;
    *reinterpret_cast<float4*>(mbp + (size_t)j * kHW) = zero4;
  }
#pragma unroll
  for (int i = 0; i < 4; ++i) {
    mfp[(size_t)lf[i] * kHW + i] = 1.0f;
    mbp[(size_t)lb[i] * kHW + i] = 1.0f;
  }
}

extern "C" void kernel_launch(void* const* d_in, const int* in_sizes, int n_in,
                              void* d_out, int out_size, void* d_ws, size_t ws_size,
                              hipStream_t stream) {
  (void)in_sizes; (void)n_in; (void)out_size; (void)d_ws; (void)ws_size;
  const float* imF = (const float*)d_in[0];
  const float* imB = (const float*)d_in[1];
  const int*   gmF = (const int*)d_in[2];
  const int*   gmB = (const int*)d_in[3];
  // d_in[4] (m_kernel) and scalar params are compile-time constants here.
  float* out = (float*)d_out;

  dim3 grid(kW / TILE, kH / TILE, kB);  // 8 x 8 x 16 = 1024 blocks
  hipLaunchKernelGGL(gtnet_fused, grid, dim3(NTHR), 0, stream,
                     imF, imB, gmF, gmB, out);
}